// AttentionWithContext_49624052138324
// MI455X (gfx1250) — compile-verified
//
#include <hip/hip_runtime.h>
#include <math.h>

typedef __attribute__((ext_vector_type(2))) float v2f;
typedef __attribute__((ext_vector_type(8))) float v8f;

#define B_   64
#define T_   2048
#define D_   256
#define C_   256
#define SPLIT 8
#define EPS_ 1e-32f

// ---------------------------------------------------------------------------
// Kernel 0: fold the dense layer into a single vector: v = W @ u, bu = b . u
// ---------------------------------------------------------------------------
__global__ void prep_kernel(const float* __restrict__ W, const float* __restrict__ b,
                            const float* __restrict__ u, float* __restrict__ v,
                            float* __restrict__ bu) {
    __shared__ float red[256];
    const int d = threadIdx.x;            // 256 threads
    float acc = 0.f;
    #pragma unroll 4
    for (int c = 0; c < C_; ++c) acc = fmaf(W[d * C_ + c], u[c], acc);
    v[d] = acc;
    red[d] = b[d] * u[d];
    __syncthreads();
    for (int s = 128; s > 0; s >>= 1) {
        if (d < s) red[d] += red[d + s];
        __syncthreads();
    }
    if (d == 0) *bu = red[0];
}

// ---------------------------------------------------------------------------
// Kernel 1: logits[b,t] = (x[b,t,:] . v + bu) / sqrt(D), via V_WMMA_F32_16X16X4_F32
// One wave computes 16 logits (16 t-rows), accumulating K=4 chunks over D=256.
// A tile 16x4:  lanes 0-15 hold {x[t0+l][d0], x[t0+l][d0+1]},
//               lanes 16-31 hold {x[t0+l-16][d0+2], x[t0+l-16][d0+3]}.
// B tile 4x16:  v[d0+k] broadcast across all N columns -> every column of D
//               holds the same 16 dot products.
// ---------------------------------------------------------------------------
__global__ void logits_kernel(const float* __restrict__ x, const float* __restrict__ v,
                              const float* __restrict__ bu_p, float* __restrict__ logits) {
    __shared__ float vs[D_];
    const int tid  = threadIdx.x;                 // 256 threads = 8 waves
    vs[tid] = v[tid];
    __syncthreads();
    const float bu = *bu_p;

    const int wave = tid >> 5;
    const int lane = tid & 31;
    const int bidx = blockIdx.x >> 4;             // 16 blocks per batch row
    const int tblk = blockIdx.x & 15;
    const int t0   = tblk * 128 + wave * 16;

    const int   koff = (lane >> 4) << 1;          // 0 for lanes 0-15, 2 for lanes 16-31
    const float* xrow = x + ((size_t)bidx * T_ + t0 + (lane & 15)) * D_;

    v8f acc = {};
    #pragma unroll 4
    for (int d0 = 0; d0 < D_; d0 += 4) {
        v2f a  = *(const v2f*)(xrow + d0 + koff);
        v2f bb = *(const v2f*)(vs   + d0 + koff);
        acc = __builtin_amdgcn_wmma_f32_16x16x4_f32(
            /*neg_a=*/false, a, /*neg_b=*/false, bb,
            /*c_mod=*/(short)0, acc, /*reuse_a=*/false, /*reuse_b=*/false);
    }

    // Column N=0 lives on lane 0 (M=0..7 in acc[0..7]) and lane 16 (M=8..15).
    if ((lane & 15) == 0) {
        float* lp = logits + (size_t)bidx * T_ + t0 + ((lane >> 4) * 8);
        #pragma unroll
        for (int j = 0; j < 8; ++j) lp[j] = (acc[j] + bu) * 0.0625f;  // 1/sqrt(256)
    }
}

// ---------------------------------------------------------------------------
// Kernel 2: padding-aware softmax over T per batch row, in place on `scores`
// (which currently holds logits). Matches reference: masked -> row min, shift
// by row max, w = mask*exp(.), normalize by (sum + EPS).
// ---------------------------------------------------------------------------
__global__ void softmax_kernel(const float* __restrict__ mask, float* __restrict__ scores) {
    __shared__ float red[256];
    const int b   = blockIdx.x;
    const int tid = threadIdx.x;                  // 256 threads, 8 elems each
    const float* mrow = mask   + (size_t)b * T_;
    float*       srow = scores + (size_t)b * T_;

    float lg[8], mk[8];
    float lmin = INFINITY;
    #pragma unroll
    for (int j = 0; j < 8; ++j) {
        const int t = tid + j * 256;
        lg[j] = srow[t];
        mk[j] = mrow[t];
        lmin  = fminf(lmin, lg[j]);
    }
    red[tid] = lmin; __syncthreads();
    for (int s = 128; s > 0; s >>= 1) { if (tid < s) red[tid] = fminf(red[tid], red[tid + s]); __syncthreads(); }
    lmin = red[0]; __syncthreads();

    float lmax = -INFINITY;
    #pragma unroll
    for (int j = 0; j < 8; ++j) {
        lg[j] = (mk[j] > 0.5f) ? lg[j] : lmin;
        lmax  = fmaxf(lmax, lg[j]);
    }
    red[tid] = lmax; __syncthreads();
    for (int s = 128; s > 0; s >>= 1) { if (tid < s) red[tid] = fmaxf(red[tid], red[tid + s]); __syncthreads(); }
    lmax = red[0]; __syncthreads();

    float lsum = 0.f;
    #pragma unroll
    for (int j = 0; j < 8; ++j) {
        lg[j] = mk[j] * expf(lg[j] - lmax);
        lsum += lg[j];
    }
    red[tid] = lsum; __syncthreads();
    for (int s = 128; s > 0; s >>= 1) { if (tid < s) red[tid] += red[tid + s]; __syncthreads(); }
    const float inv = 1.0f / (red[0] + EPS_);
    #pragma unroll
    for (int j = 0; j < 8; ++j) srow[tid + j * 256] = lg[j] * inv;
}

// ---------------------------------------------------------------------------
// Kernel 3: partial weighted sum over a T-chunk: part[b,s,d] = sum_t s_t * x[b,t,d]
// ---------------------------------------------------------------------------
__global__ void wsum_kernel(const float* __restrict__ x, const float* __restrict__ scores,
                            float* __restrict__ part) {
    __shared__ float sS[T_ / SPLIT];              // 256 scores for this chunk
    const int b = blockIdx.x;
    const int s = blockIdx.y;
    const int d = threadIdx.x;                    // 256 threads
    const size_t tbase = (size_t)b * T_ + (size_t)s * (T_ / SPLIT);
    sS[d] = scores[tbase + d];
    __syncthreads();

    const float* xcol = x + tbase * D_ + d;
    float acc = 0.f;
    #pragma unroll 4
    for (int t = 0; t < T_ / SPLIT; ++t)
        acc = fmaf(sS[t], xcol[(size_t)t * D_], acc);

    part[((size_t)b * SPLIT + s) * D_ + d] = acc;
}

// ---------------------------------------------------------------------------
// Kernel 4: fold SPLIT partials into output[b,d]
// ---------------------------------------------------------------------------
__global__ void reduce_kernel(const float* __restrict__ part, float* __restrict__ out) {
    const int idx = blockIdx.x * 256 + threadIdx.x;  // b*D + d
    const int b = idx >> 8;
    const int d = idx & 255;
    float acc = 0.f;
    #pragma unroll
    for (int s = 0; s < SPLIT; ++s) acc += part[((size_t)b * SPLIT + s) * D_ + d];
    out[idx] = acc;
}

// ---------------------------------------------------------------------------
extern "C" void kernel_launch(void* const* d_in, const int* in_sizes, int n_in,
                              void* d_out, int out_size, void* d_ws, size_t ws_size,
                              hipStream_t stream) {
    const float* x    = (const float*)d_in[0];   // [B,T,D]
    const float* mask = (const float*)d_in[1];   // [B,T]
    const float* W    = (const float*)d_in[2];   // [D,C]
    const float* bb   = (const float*)d_in[3];   // [C]
    const float* u    = (const float*)d_in[4];   // [C]

    float* out    = (float*)d_out;               // [B,D] = 16384 floats
    float* scores = out + B_ * D_;               // [B,T] = 131072 floats

    float* v    = (float*)d_ws;                  // 256 floats
    float* bu   = v + D_;                        // 1 float
    float* part = v + 512;                       // [B, SPLIT, D] = 131072 floats

    prep_kernel  <<<1,               256, 0, stream>>>(W, bb, u, v, bu);
    logits_kernel<<<B_ * (T_ / 128), 256, 0, stream>>>(x, v, bu, scores);
    softmax_kernel<<<B_,             256, 0, stream>>>(mask, scores);
    dim3 g(B_, SPLIT);
    wsum_kernel  <<<g,               256, 0, stream>>>(x, scores, part);
    reduce_kernel<<<B_ * D_ / 256,   256, 0, stream>>>(part, out);
}